// GCNLayer_33449205301469
// MI455X (gfx1250) — compile-verified
//
#include <hip/hip_runtime.h>

#define N_NODES 100000
#define N_EDGES 1600000
#define IN_F    48
#define OUT_F   48
#define NTILES  (N_NODES / 16)   // 6250, exact

typedef float v2f __attribute__((ext_vector_type(2)));
typedef float v8f __attribute__((ext_vector_type(8)));

// -------- kernel 1: zero the accumulator (d_out) and degree array ----------
__global__ void gcn_zero_kernel(float* __restrict__ acc, int* __restrict__ deg) {
    int i = blockIdx.x * blockDim.x + threadIdx.x;
    if (i < N_NODES * OUT_F) acc[i] = 0.0f;
    if (i < N_NODES) deg[i] = 0;
}

// -------- kernel 2: degree via int atomics on row targets ------------------
__global__ void gcn_deg_kernel(const int* __restrict__ row, int* __restrict__ deg) {
    int e = blockIdx.x * blockDim.x + threadIdx.x;
    if (e < N_EDGES) atomicAdd(&deg[row[e]], 1);
}

// -------- kernel 3: deg^{-1/2} ---------------------------------------------
__global__ void gcn_dis_kernel(const int* __restrict__ deg, float* __restrict__ dis) {
    int i = blockIdx.x * blockDim.x + threadIdx.x;
    if (i < N_NODES) {
        int d = deg[i];
        dis[i] = (d > 0) ? rsqrtf((float)d) : 0.0f;
    }
}

// -------- kernel 4: xW = x @ W^T via V_WMMA_F32_16X16X4_F32 ----------------
// One wave handles a 16-node tile x 48 outputs. 10 waves/block, 625 blocks
// covers exactly 6250 tiles -> EXEC is all-ones for every WMMA (required).
__global__ void __launch_bounds__(320)
gcn_gemm_kernel(const float* __restrict__ x, const float* __restrict__ W,
                float* __restrict__ xW) {
    const int wave = threadIdx.x >> 5;
    const int lane = threadIdx.x & 31;
    const int tile = blockIdx.x * 10 + wave;       // < 6250 by construction
    const int m0   = tile * 16;
    const int half = lane >> 4;                    // 0: K lo pair, 1: K hi pair
    const int lm   = lane & 15;

    // Preload A: row (m0+lm) of x, K split 2*half across lane halves, 12 K-steps.
    v2f a[12];
    const float* xrow = x + (size_t)(m0 + lm) * IN_F + 2 * half;
#pragma unroll
    for (int k = 0; k < 12; ++k)
        a[k] = *(const v2f*)(xrow + 4 * k);

    v8f acc[3] = {};                               // 3 output tiles of 16 cols
#pragma unroll
    for (int n = 0; n < 3; ++n) {
        // B is W^T tile: element (k, col) = W[col][k]; contiguous along k.
        const float* wrow = W + (size_t)(16 * n + lm) * IN_F + 2 * half;
#pragma unroll
        for (int k = 0; k < 12; ++k) {
            v2f b = *(const v2f*)(wrow + 4 * k);
            acc[n] = __builtin_amdgcn_wmma_f32_16x16x4_f32(
                false, a[k], false, b, (short)0, acc[n], false, false);
        }
    }

    // D layout: VGPR v -> M = v + 8*half, N = lm (within the 16-col tile)
#pragma unroll
    for (int n = 0; n < 3; ++n) {
        float* o = xW + (size_t)(m0 + 8 * half) * OUT_F + 16 * n + lm;
#pragma unroll
        for (int v = 0; v < 8; ++v)
            o[(size_t)v * OUT_F] = acc[n][v];
    }
}

// -------- kernel 5: edge scatter, one thread per (edge, feature) -----------
__global__ void gcn_scatter_kernel(const int* __restrict__ ei,
                                   const float* __restrict__ dis,
                                   const float* __restrict__ xW,
                                   float* __restrict__ acc) {
    int idx = blockIdx.x * blockDim.x + threadIdx.x;   // < 76.8M, fits int
    if (idx >= N_EDGES * OUT_F) return;
    int e = idx / OUT_F;
    int f = idx - e * OUT_F;
    int r = ei[e];               // target (segment) node
    int c = ei[N_EDGES + e];     // source node
    float norm = dis[r] * dis[c];
    atomicAdd(&acc[(size_t)r * OUT_F + f], norm * xW[(size_t)c * OUT_F + f]);
}

// -------- kernel 6: out = relu(acc + bias), in place on d_out --------------
__global__ void gcn_bias_relu_kernel(float* __restrict__ acc,
                                     const float* __restrict__ b) {
    int i = blockIdx.x * blockDim.x + threadIdx.x;
    if (i < N_NODES * OUT_F) {
        int f = i % OUT_F;
        float v = acc[i] + b[f];
        acc[i] = v > 0.0f ? v : 0.0f;
    }
}

extern "C" void kernel_launch(void* const* d_in, const int* in_sizes, int n_in,
                              void* d_out, int out_size, void* d_ws, size_t ws_size,
                              hipStream_t stream) {
    const float* x  = (const float*)d_in[0];   // [N, 48]
    const int*   ei = (const int*)d_in[1];     // [2, E]
    const float* W  = (const float*)d_in[2];   // [48, 48]
    const float* b  = (const float*)d_in[3];   // [48]
    float* out = (float*)d_out;                // [N, 48], used as accumulator

    // ws layout (all 16B aligned): deg | dis | xW
    char* ws = (char*)d_ws;
    int*   deg = (int*)ws;                                   // N ints
    float* dis = (float*)(ws + (size_t)N_NODES * 4);         // N floats
    float* xW  = (float*)(ws + (size_t)N_NODES * 8);         // N*48 floats

    const int NF = N_NODES * OUT_F;     // 4.8M
    gcn_zero_kernel<<<(NF + 255) / 256, 256, 0, stream>>>(out, deg);
    gcn_deg_kernel<<<(N_EDGES + 255) / 256, 256, 0, stream>>>(ei, deg);
    gcn_dis_kernel<<<(N_NODES + 255) / 256, 256, 0, stream>>>(deg, dis);
    gcn_gemm_kernel<<<NTILES / 10, 320, 0, stream>>>(x, W, xW);
    gcn_scatter_kernel<<<(N_EDGES * OUT_F + 255) / 256, 256, 0, stream>>>(ei, dis, xW, out);
    gcn_bias_relu_kernel<<<(NF + 255) / 256, 256, 0, stream>>>(out, b);
}